// UniformLoss_40561671144092
// MI455X (gfx1250) — compile-verified
//
#include <hip/hip_runtime.h>
#include <hip/hip_bf16.h>
#include <math.h>

#define BDIM 8192
#define DDIM 128
#define EPSV 1e-8f

typedef __attribute__((ext_vector_type(2))) float v2f;
typedef __attribute__((ext_vector_type(8))) float v8f;

// ---------------------------------------------------------------------------
// Kernel 1: row L2 norms for A and B.  One wave32 per row (D=128 -> 4 f32/lane)
// ---------------------------------------------------------------------------
__global__ void norms_kernel(const float* __restrict__ A,
                             const float* __restrict__ Bf,
                             float* __restrict__ ws) {
    const int gwave = (int)((blockIdx.x * blockDim.x + threadIdx.x) >> 5);
    const int lane  = (int)(threadIdx.x & 31);
    const float* src = (gwave < BDIM) ? A : Bf;
    const int row    = (gwave < BDIM) ? gwave : gwave - BDIM;
    const float4 v = *(const float4*)(src + (size_t)row * DDIM + lane * 4);
    float s = v.x * v.x + v.y * v.y + v.z * v.z + v.w * v.w;
    #pragma unroll
    for (int off = 16; off > 0; off >>= 1) s += __shfl_xor(s, off, 32);
    if (lane == 0) ws[gwave] = sqrtf(s);
}

// ---------------------------------------------------------------------------
// Kernel 2: cos[i][j] = (A @ B^T)[i][j] / max(na[i]*nb[j], eps)
// Wave computes 32x32 tile via 2x2 grid of 16x16 f32 WMMA accumulators.
// Block = 8 waves (wave32) -> 64 rows x 128 cols per block.
// ---------------------------------------------------------------------------
__global__ void __launch_bounds__(256)
gemm_cos_kernel(const float* __restrict__ A, const float* __restrict__ Bf,
                const float* __restrict__ na, const float* __restrict__ nb,
                float* __restrict__ cosOut) {
    const int lane = (int)(threadIdx.x & 31);
    const int wid  = (int)(threadIdx.x >> 5);
    const int wrow = wid & 1;   // 2 wave-rows
    const int wcol = wid >> 1;  // 4 wave-cols
    const int row0 = (int)blockIdx.y * 64  + wrow * 32;
    const int col0 = (int)blockIdx.x * 128 + wcol * 32;

    // A-frag (16x4 f32): lanes 0-15 hold M=lane, K={0,1}; lanes 16-31 M=lane-16, K={2,3}
    const int m  = lane & 15;
    const int kk = (lane >> 4) << 1;  // 0 or 2

    const float* a0p = A  + (size_t)(row0      + m) * DDIM + kk;
    const float* a1p = A  + (size_t)(row0 + 16 + m) * DDIM + kk;
    const float* b0p = Bf + (size_t)(col0      + m) * DDIM + kk;
    const float* b1p = Bf + (size_t)(col0 + 16 + m) * DDIM + kk;

    v8f c00 = {}, c01 = {}, c10 = {}, c11 = {};

    #pragma unroll 8
    for (int s = 0; s < DDIM / 4; ++s) {
        const v2f a0 = *(const v2f*)(a0p + 4 * s);
        const v2f a1 = *(const v2f*)(a1p + 4 * s);
        const v2f b0 = *(const v2f*)(b0p + 4 * s);
        const v2f b1 = *(const v2f*)(b1p + 4 * s);
        // 8 args: (neg_a, A, neg_b, B, c_mod, C, reuse_a, reuse_b)
        c00 = __builtin_amdgcn_wmma_f32_16x16x4_f32(false, a0, false, b0, (short)0, c00, false, false);
        c01 = __builtin_amdgcn_wmma_f32_16x16x4_f32(false, a0, false, b1, (short)0, c01, false, false);
        c10 = __builtin_amdgcn_wmma_f32_16x16x4_f32(false, a1, false, b0, (short)0, c10, false, false);
        c11 = __builtin_amdgcn_wmma_f32_16x16x4_f32(false, a1, false, b1, (short)0, c11, false, false);
    }

    // C/D layout: VGPR g -> M = g + (lane>=16 ? 8 : 0), N = lane & 15
    const int n    = lane & 15;
    const int moff = (lane >> 4) * 8;
    const float nb0 = nb[col0 + n];
    const float nb1 = nb[col0 + 16 + n];

    #pragma unroll
    for (int g = 0; g < 8; ++g) {
        const int r0 = row0 + moff + g;
        const int r1 = row0 + 16 + moff + g;
        const float na0 = na[r0];
        const float na1 = na[r1];
        cosOut[(size_t)r0 * BDIM + col0 + n]      = c00[g] / fmaxf(na0 * nb0, EPSV);
        cosOut[(size_t)r0 * BDIM + col0 + 16 + n] = c01[g] / fmaxf(na0 * nb1, EPSV);
        cosOut[(size_t)r1 * BDIM + col0 + n]      = c10[g] / fmaxf(na1 * nb0, EPSV);
        cosOut[(size_t)r1 * BDIM + col0 + 16 + n] = c11[g] / fmaxf(na1 * nb1, EPSV);
    }
}

// ---------------------------------------------------------------------------
// Kernel 3: per-row log-softmax + label-masked NLL partial.
// One 256-thread block per row; row cached in 32KB LDS (WGP has 320KB).
// ---------------------------------------------------------------------------
__global__ void __launch_bounds__(256)
row_loss_kernel(const float* __restrict__ cosM, const int* __restrict__ labels,
                float* __restrict__ partial) {
    __shared__ float rowbuf[BDIM];
    __shared__ float red[8];

    const int i    = (int)blockIdx.x;
    const int t    = (int)threadIdx.x;
    const int lane = t & 31;
    const int wid  = t >> 5;
    const float* row = cosM + (size_t)i * BDIM;

    // Pass 1: load row into LDS, find max
    float mx = -INFINITY;
    for (int j = t; j < BDIM; j += 256) {
        const float v = row[j];
        rowbuf[j] = v;
        mx = fmaxf(mx, v);
    }
    #pragma unroll
    for (int off = 16; off > 0; off >>= 1) mx = fmaxf(mx, __shfl_xor(mx, off, 32));
    if (lane == 0) red[wid] = mx;
    __syncthreads();
    float rmax = red[0];
    #pragma unroll
    for (int w = 1; w < 8; ++w) rmax = fmaxf(rmax, red[w]);
    __syncthreads();

    // Pass 2: sum of exp(x - max)
    float se = 0.f;
    for (int j = t; j < BDIM; j += 256) se += __expf(rowbuf[j] - rmax);
    #pragma unroll
    for (int off = 16; off > 0; off >>= 1) se += __shfl_xor(se, off, 32);
    if (lane == 0) red[wid] = se;
    __syncthreads();
    float tot = 0.f;
    #pragma unroll
    for (int w = 0; w < 8; ++w) tot += red[w];
    const float lse = __logf(tot) + rmax;
    __syncthreads();

    // Pass 3: masked sum of logp
    const int li = labels[i];
    float acc = 0.f;
    for (int j = t; j < BDIM; j += 256)
        if (labels[j] == li) acc += rowbuf[j] - lse;
    #pragma unroll
    for (int off = 16; off > 0; off >>= 1) acc += __shfl_xor(acc, off, 32);
    if (lane == 0) red[wid] = acc;
    __syncthreads();
    if (t == 0) {
        float s = 0.f;
        #pragma unroll
        for (int w = 0; w < 8; ++w) s += red[w];
        partial[i] = s;
    }
}

// ---------------------------------------------------------------------------
// Kernel 4: reduce per-row partials -> loss scalar
// ---------------------------------------------------------------------------
__global__ void __launch_bounds__(256)
final_loss_kernel(const float* __restrict__ partial, float* __restrict__ out) {
    __shared__ float red[8];
    const int t = (int)threadIdx.x;
    const int lane = t & 31, wid = t >> 5;
    float s = 0.f;
    for (int j = t; j < BDIM; j += 256) s += partial[j];
    #pragma unroll
    for (int off = 16; off > 0; off >>= 1) s += __shfl_xor(s, off, 32);
    if (lane == 0) red[wid] = s;
    __syncthreads();
    if (t == 0) {
        float tt = 0.f;
        #pragma unroll
        for (int w = 0; w < 8; ++w) tt += red[w];
        out[0] = -tt / ((float)BDIM * (float)BDIM);
    }
}

// ---------------------------------------------------------------------------
extern "C" void kernel_launch(void* const* d_in, const int* in_sizes, int n_in,
                              void* d_out, int out_size, void* d_ws, size_t ws_size,
                              hipStream_t stream) {
    const int*   labels = (const int*)d_in[0];
    const float* fa     = (const float*)d_in[1];
    const float* fb     = (const float*)d_in[2];

    float* out  = (float*)d_out;           // out[0] = loss, out+1 = cos [B,B]
    float* cosM = out + 1;

    float* na      = (float*)d_ws;         // [B]
    float* nb      = na + BDIM;            // [B]
    float* partial = nb + BDIM;            // [B]

    // 1) norms: 2*B rows, one wave per row, 8 waves per block
    norms_kernel<<<dim3(2 * BDIM / 8), dim3(256), 0, stream>>>(fa, fb, (float*)d_ws);

    // 2) WMMA GEMM + cosine scale: block tile 64 rows x 128 cols
    gemm_cos_kernel<<<dim3(BDIM / 128, BDIM / 64), dim3(256), 0, stream>>>(
        fa, fb, na, nb, cosM);

    // 3) per-row log-softmax + masked NLL partials
    row_loss_kernel<<<dim3(BDIM), dim3(256), 0, stream>>>(cosM, labels, partial);

    // 4) final reduction -> loss
    final_loss_kernel<<<dim3(1), dim3(256), 0, stream>>>(partial, out);
}